// ESN_81432579932375
// MI455X (gfx1250) — compile-verified
//
#include <hip/hip_runtime.h>
#include <hip/hip_bf16.h>
#include <math.h>

// ---------------------------------------------------------------------------
// ESN on MI455X (gfx1250):
//   1) proj  = u @ W_in^T           : fp32 WMMA GEMM (V_WMMA_F32_16X16X4_F32)
//   2) scan  : persistent kernel, W resident in LDS (128 WGs x 16 rows),
//              per-step state broadcast via GLOBAL_LOAD_ASYNC_TO_LDS_B128,
//              grid-wide release/acquire barrier per timestep
//   3) out   = states @ W_out^T + b : fp32 WMMA GEMM (+ global_prefetch)
// ---------------------------------------------------------------------------

typedef __attribute__((ext_vector_type(2))) float v2f;
typedef __attribute__((ext_vector_type(8))) float v8f;

#define T_STEPS 8192
#define D_IN    128
#define N_RES   2048
#define O_OUT   128

#define LEAK    0.3f

// scan partitioning
#define SCAN_WGS   128
#define ROWS_PER_WG 16            // 128 * 16 = 2048 rows
#define SPAD       2064           // padded LDS row stride (floats): 2064 % 64 == 16 -> bank-skewed
#define SCAN_LDS_BYTES ((ROWS_PER_WG * SPAD + N_RES) * sizeof(float))

// ---------------------------------------------------------------------------
// Async global -> LDS copy (CDNA5 GLOBAL_LOAD_ASYNC_TO_LDS_B128, ASYNCcnt),
// with a portable fallback if this toolchain lacks the builtin.
// Builtin signature (from clang diagnostic): param0 is
//   'int __attribute__((vector_size(16))) __device__ *'  (addrspace(1), non-const)
// ---------------------------------------------------------------------------
#if __has_builtin(__builtin_amdgcn_global_load_async_to_lds_b128)
#define HAVE_ASYNC_LDS 1
#else
#define HAVE_ASYNC_LDS 0
#endif

typedef int vec4i __attribute__((vector_size(4 * sizeof(int))));
typedef __attribute__((address_space(1))) vec4i* as1_v4i_ptr;   // global
typedef __attribute__((address_space(3))) vec4i* as3_v4i_ptr;   // LDS

__device__ __forceinline__ void lds_copy_b128(void* lds_dst, const void* gsrc)
{
#if HAVE_ASYNC_LDS
    __builtin_amdgcn_global_load_async_to_lds_b128(
        (as1_v4i_ptr)gsrc, (as3_v4i_ptr)lds_dst, 0, 0);
#else
    *(float4*)lds_dst = *(const float4*)gsrc;
#endif
}

__device__ __forceinline__ void lds_copy_wait()
{
#if HAVE_ASYNC_LDS
#if __has_builtin(__builtin_amdgcn_s_wait_asynccnt)
    __builtin_amdgcn_s_wait_asynccnt(0);
#else
    asm volatile("s_wait_asynccnt 0" ::: "memory");
#endif
#endif
}

// ---------------------------------------------------------------------------
// Kernel 1: proj[T,N] = u[T,D] @ W_in[N,D]^T   (fp32 WMMA, K-step 4)
// one wave per 16x16 output tile, 8 waves per block
// ---------------------------------------------------------------------------
__global__ __launch_bounds__(256) void esn_proj_gemm(
    const float* __restrict__ u, const float* __restrict__ Win,
    float* __restrict__ P)
{
    const int wave = threadIdx.x >> 5;
    const int lane = threadIdx.x & 31;
    const int tile = blockIdx.x * 8 + wave;
    const int ntn  = N_RES / 16;              // 128 tiles along N
    const int tm   = tile / ntn;
    const int tn   = tile - tm * ntn;
    const int m0   = tm * 16;
    const int n0   = tn * 16;

    const int half = lane >> 4;               // 0: K pair {0,1}, 1: K pair {2,3}
    const int idx  = lane & 15;               // A: row M, B: col N

    const float* ap = u   + (size_t)(m0 + idx) * D_IN + 2 * half;
    const float* bp = Win + (size_t)(n0 + idx) * D_IN + 2 * half;

    v8f acc = {};
#pragma unroll 8
    for (int k = 0; k < D_IN; k += 4) {
        v2f a, b;
        a.x = ap[0]; a.y = ap[1];
        b.x = bp[0]; b.y = bp[1];
        ap += 4; bp += 4;
        acc = __builtin_amdgcn_wmma_f32_16x16x4_f32(
            false, a, false, b, (short)0, acc, false, false);
    }

    // C/D layout: VGPR r -> row (r + 8*half), col idx
    float* outp = P + (size_t)(m0 + 8 * half) * N_RES + n0 + idx;
#pragma unroll
    for (int r = 0; r < 8; ++r)
        outp[(size_t)r * N_RES] = acc[r];
}

// ---------------------------------------------------------------------------
// Kernel 2: persistent scan. Each WG owns 16 rows of W, held in LDS.
// Per step: s_next[row] = 0.7*s[row] + 0.3*tanh(P[t,row] + dot(W[row,:], s))
// states are written back over P in place. Grid sync via global atomic.
// ---------------------------------------------------------------------------
extern __shared__ float scan_lds[];

__global__ __launch_bounds__(256) void esn_scan(
    const float* __restrict__ W,
    float* __restrict__ P,              // in: proj, out: states (in place)
    float* __restrict__ sbuf,           // 2 * N_RES (double buffered state)
    unsigned int* __restrict__ bar)
{
    float* wl = scan_lds;                       // ROWS_PER_WG * SPAD
    float* sl = scan_lds + ROWS_PER_WG * SPAD;  // N_RES (current state copy)

    const int tid = threadIdx.x;
    const int wg  = blockIdx.x;

    // --- preload this WG's 16 rows of W into (bank-padded) LDS (async DMA) ---
    {
        const float* wgp = W + (size_t)wg * ROWS_PER_WG * N_RES;
        for (int i = tid; i < ROWS_PER_WG * (N_RES / 4); i += 256) {
            const int r  = i >> 9;          // / (N_RES/4)
            const int c4 = i & 511;
            lds_copy_b128(wl + r * SPAD + c4 * 4,
                          wgp + (size_t)r * N_RES + c4 * 4);
        }
        lds_copy_wait();
    }
    __syncthreads();

    const int row  = tid >> 4;          // 0..15  (16 threads per row)
    const int sub  = tid & 15;          // 0..15
    const int grow = wg * ROWS_PER_WG + row;

    float* s0 = sbuf;
    float* s1 = sbuf + N_RES;

    for (int t = 0; t < T_STEPS; ++t) {
        float* scur = (t & 1) ? s1 : s0;
        float* snxt = (t & 1) ? s0 : s1;

        // broadcast-load the full state vector into LDS (8 KB, L2-resident):
        // direct L2 -> LDS, no VGPR round-trip, tracked by ASYNCcnt
        for (int i = tid; i < N_RES / 4; i += 256)
            lds_copy_b128(sl + i * 4, scur + i * 4);
        lds_copy_wait();
        __syncthreads();

        // 16 threads per row, float4 strided: conflict-free LDS traffic
        float acc = 0.0f;
        const float* wr = wl + row * SPAD;
#pragma unroll 8
        for (int i = 0; i < 32; ++i) {
            const int c = sub * 4 + i * 64;
            float4 w4 = *(const float4*)(wr + c);
            float4 s4 = *(const float4*)(sl + c);
            acc += w4.x * s4.x + w4.y * s4.y + w4.z * s4.z + w4.w * s4.w;
        }
        // reduce across the 16 lanes of this row (lanes stay in one half-wave)
        acc += __shfl_xor(acc, 8, 32);
        acc += __shfl_xor(acc, 4, 32);
        acc += __shfl_xor(acc, 2, 32);
        acc += __shfl_xor(acc, 1, 32);

        if (sub == 0) {
            const float z  = P[(size_t)t * N_RES + grow] + acc;
            const float sn = (1.0f - LEAK) * sl[grow] + LEAK * tanhf(z);
            snxt[grow] = sn;
            P[(size_t)t * N_RES + grow] = sn;   // record state
        }

        // ---- grid-wide barrier: release my writes, arrive, wait, acquire ----
        __threadfence();                        // agent-scope release of stores
        __syncthreads();
        if (tid == 0) {
            __hip_atomic_fetch_add(bar, 1u, __ATOMIC_ACQ_REL,
                                   __HIP_MEMORY_SCOPE_AGENT);
            const unsigned int tgt = (unsigned int)(t + 1) * SCAN_WGS;
            while (__hip_atomic_load(bar, __ATOMIC_ACQUIRE,
                                     __HIP_MEMORY_SCOPE_AGENT) < tgt)
                __builtin_amdgcn_s_sleep(2);
        }
        __syncthreads();
        __threadfence();                        // acquire: invalidate stale lines
    }
}

// ---------------------------------------------------------------------------
// Kernel 3: out[T,O] = states[T,N] @ W_out[O,N]^T + b_out  (fp32 WMMA)
// K blocked by 64 with global_prefetch of the next block
// ---------------------------------------------------------------------------
__global__ __launch_bounds__(256) void esn_readout_gemm(
    const float* __restrict__ S, const float* __restrict__ Wout,
    const float* __restrict__ bout, float* __restrict__ out)
{
    const int wave = threadIdx.x >> 5;
    const int lane = threadIdx.x & 31;
    const int tile = blockIdx.x * 8 + wave;     // ntn = O/16 = 8
    const int tm   = tile >> 3;
    const int tn   = tile & 7;
    const int m0   = tm * 16;
    const int n0   = tn * 16;

    const int half = lane >> 4;
    const int idx  = lane & 15;

    const float* ap = S    + (size_t)(m0 + idx) * N_RES + 2 * half;
    const float* bp = Wout + (size_t)(n0 + idx) * N_RES + 2 * half;

    v8f acc = {};
    for (int kk = 0; kk < N_RES; kk += 64) {
        // prefetch the next 64-float K block of this row (global_prefetch_b8)
        __builtin_prefetch(ap + 64, 0, 1);
        __builtin_prefetch(bp + 64, 0, 1);
#pragma unroll
        for (int k = 0; k < 64; k += 4) {
            v2f a, b;
            a.x = ap[0]; a.y = ap[1];
            b.x = bp[0]; b.y = bp[1];
            ap += 4; bp += 4;
            acc = __builtin_amdgcn_wmma_f32_16x16x4_f32(
                false, a, false, b, (short)0, acc, false, false);
        }
    }

    const float bias = bout[n0 + idx];
    float* outp = out + (size_t)(m0 + 8 * half) * O_OUT + n0 + idx;
#pragma unroll
    for (int r = 0; r < 8; ++r)
        outp[(size_t)r * O_OUT] = acc[r] + bias;
}

// ---------------------------------------------------------------------------
// init: zero the t=0 state buffer and the barrier counter (ws is poisoned)
// ---------------------------------------------------------------------------
__global__ void esn_init(float* __restrict__ sbuf, unsigned int* __restrict__ bar)
{
    const int i = blockIdx.x * blockDim.x + threadIdx.x;
    if (i < N_RES) sbuf[i] = 0.0f;
    if (i == 0)    *bar    = 0u;
}

// ---------------------------------------------------------------------------
extern "C" void kernel_launch(void* const* d_in, const int* in_sizes, int n_in,
                              void* d_out, int out_size, void* d_ws, size_t ws_size,
                              hipStream_t stream)
{
    const float* u    = (const float*)d_in[0];   // [T, D]
    const float* Win  = (const float*)d_in[1];   // [N, D]
    const float* W    = (const float*)d_in[2];   // [N, N]
    const float* Wout = (const float*)d_in[3];   // [O, N]
    const float* bout = (const float*)d_in[4];   // [O]
    float* out = (float*)d_out;                  // [T, O]

    char* ws = (char*)d_ws;
    float* P = (float*)ws;                                        // T*N fp32 (proj -> states)
    float* sbuf = (float*)(ws + (size_t)T_STEPS * N_RES * 4);     // 2*N fp32
    unsigned int* bar =
        (unsigned int*)(ws + (size_t)T_STEPS * N_RES * 4 + (size_t)2 * N_RES * 4);

    // allow >64KB dynamic LDS for the persistent scan kernel (deterministic,
    // not a stream op -> safe under graph capture)
    (void)hipFuncSetAttribute((const void*)esn_scan,
                              hipFuncAttributeMaxDynamicSharedMemorySize,
                              (int)SCAN_LDS_BYTES);

    esn_init<<<(N_RES + 255) / 256, 256, 0, stream>>>(sbuf, bar);

    // proj: (T/16)*(N/16) = 512*128 = 65536 wave-tiles, 8 waves/block
    esn_proj_gemm<<<65536 / 8, 256, 0, stream>>>(u, Win, P);

    // sequential reservoir scan (persistent grid)
    esn_scan<<<SCAN_WGS, 256, SCAN_LDS_BYTES, stream>>>(W, P, sbuf, bar);

    // readout: (T/16)*(O/16) = 512*8 = 4096 wave-tiles, 8 waves/block
    esn_readout_gemm<<<4096 / 8, 256, 0, stream>>>(P, Wout, bout, out);
}